// NodeEncoder_28613072126470
// MI455X (gfx1250) — compile-verified
//
#include <hip/hip_runtime.h>
#include <hip/hip_bf16.h>

// Temporal graph scatter-mean:
//   mask_e = (edge_time[e] <= seed_time[dst]) && (edge_time[e] > seed_time[dst] - 100)
//   msum[dst] += x[src] * mask ; cnt[dst] += mask
//   out = x + msum / max(cnt, 1)
//
// Phase 1: zero accumulator (d_out) + counts (d_ws)
// Phase 2: wave-per-edge masked gather + float atomic scatter (memory bound)
// Phase 3: WMMA epilogue: out_tile = X_tile + diag(1/max(cnt,1)) * Msum_tile
//          via 4 chained V_WMMA_F32_16X16X4_F32 (exact in f32: shifted-diagonal A).

typedef __attribute__((ext_vector_type(2))) float v2f;
typedef __attribute__((ext_vector_type(8))) float v8f;

#define FEAT 128
#define TWIN 100

__global__ void edge_scatter_kernel(const float* __restrict__ x,
                                    const long long* __restrict__ edge_index,
                                    const int* __restrict__ edge_time,
                                    const int* __restrict__ seed_time,
                                    float* __restrict__ msum,
                                    int* __restrict__ cnt,
                                    long long E) {
    // 32 lanes per edge; lane handles 4 consecutive features (float4).
    long long i = (long long)blockIdx.x * blockDim.x + threadIdx.x;
    long long total = E << 5;
    if (i >= total) return;
    long long e = i >> 5;
    int lane = (int)(i & 31);

    long long src = edge_index[e];
    long long dst = edge_index[E + e];
    int t  = edge_time[e];
    int s0 = seed_time[dst];
    bool inwin = (t <= s0) && (t > s0 - TWIN);
    if (!inwin) return;

    const float4 v = *(const float4*)(x + src * FEAT + lane * 4);
    float* o = msum + dst * FEAT + lane * 4;
    atomicAdd(o + 0, v.x);
    atomicAdd(o + 1, v.y);
    atomicAdd(o + 2, v.z);
    atomicAdd(o + 3, v.w);
    if (lane == 0) atomicAdd(cnt + dst, 1);
}

// One wave per 16x16 tile. ntiles = (N/16) * (FEAT/16).
__global__ void finalize_wmma_kernel(const float* __restrict__ x,
                                     float* __restrict__ out,    // in: msum, out: result
                                     const int* __restrict__ cnt,
                                     int ntiles) {
    int wave = blockIdx.x * (blockDim.x >> 5) + (threadIdx.x >> 5);
    if (wave >= ntiles) return;
    int lane = threadIdx.x & 31;
    int ln15 = lane & 15;     // = N column for B/C/D, = M row for A
    int hf   = lane >> 4;     // lane half selects K/M offset per ISA layouts

    int node_blk = wave >> 3;         // FEAT/16 == 8 tiles per node block
    int feat_blk = wave & 7;
    size_t base = (size_t)node_blk * 16 * FEAT + (size_t)feat_blk * 16 + ln15;

    const float* xp = x + base;
    float*       op = out + base;

    // C tile: x[16x16] in C/D layout (vgpr r -> row M = r + 8*half, col = ln15)
    v8f acc;
#pragma unroll
    for (int r = 0; r < 8; ++r)
        acc[r] = xp[(size_t)(r + 8 * hf) * FEAT];

    // Per-row scale s_m = 1 / max(cnt, 1); this lane's A-row is m = ln15.
    int cv = cnt[node_blk * 16 + ln15];
    float s = 1.0f / (float)(cv > 1 ? cv : 1);

    // 4 chained WMMAs: chunk c scales rows 4c..4c+3 exactly; others get +0.
#pragma unroll
    for (int c = 0; c < 4; ++c) {
        v2f a, b;
#pragma unroll
        for (int r = 0; r < 2; ++r) {
            int k = r + 2 * hf;                       // K index per A/B layout
            a[r] = (ln15 == 4 * c + k) ? s : 0.0f;    // shifted diagonal of scales
            b[r] = op[(size_t)(4 * c + k) * FEAT];    // rows 4c..4c+3 of msum tile
        }
        acc = __builtin_amdgcn_wmma_f32_16x16x4_f32(
            /*neg_a=*/false, a, /*neg_b=*/false, b,
            /*c_mod=*/(short)0, acc, /*reuse_a=*/false, /*reuse_b=*/false);
    }

#pragma unroll
    for (int r = 0; r < 8; ++r)
        op[(size_t)(r + 8 * hf) * FEAT] = acc[r];
}

// Scalar tail for N % 16 != 0 (not hit for N = 100000, kept for generality).
__global__ void finalize_rem_kernel(const float* __restrict__ x,
                                    float* __restrict__ out,
                                    const int* __restrict__ cnt,
                                    int start, int N) {
    int total = (N - start) * FEAT;
    int i = blockIdx.x * blockDim.x + threadIdx.x;
    if (i >= total) return;
    int n = start + i / FEAT;
    int f = i % FEAT;
    int cv = cnt[n];
    float s = 1.0f / (float)(cv > 1 ? cv : 1);
    size_t idx = (size_t)n * FEAT + f;
    out[idx] = x[idx] + out[idx] * s;
}

extern "C" void kernel_launch(void* const* d_in, const int* in_sizes, int n_in,
                              void* d_out, int out_size, void* d_ws, size_t ws_size,
                              hipStream_t stream) {
    const float*     x  = (const float*)d_in[0];      // [N, 128] f32
    const long long* ei = (const long long*)d_in[1];  // [2, E] i64
    const int*       et = (const int*)d_in[2];        // [E] i32
    const int*       st = (const int*)d_in[3];        // [N] i32
    float* out = (float*)d_out;                       // [N, 128] f32
    int* cnt = (int*)d_ws;                            // [N] i32 scratch

    long long E = in_sizes[2];
    int N = in_sizes[3];

    // Phase 1: zero accumulator + counts (async memsets are graph-capturable).
    hipMemsetAsync(out, 0, (size_t)N * FEAT * sizeof(float), stream);
    hipMemsetAsync(cnt, 0, (size_t)N * sizeof(int), stream);

    // Phase 2: masked gather + atomic scatter, wave per edge.
    {
        long long total = E << 5;
        int blocks = (int)((total + 255) / 256);
        edge_scatter_kernel<<<blocks, 256, 0, stream>>>(x, ei, et, st, out, cnt, E);
    }

    // Phase 3: WMMA row-scale epilogue.
    int nblk16 = N / 16;
    int ntiles = nblk16 * (FEAT / 16);
    if (ntiles > 0) {
        int waves_per_block = 8;  // 256 threads
        int blocks = (ntiles + waves_per_block - 1) / waves_per_block;
        finalize_wmma_kernel<<<blocks, 256, 0, stream>>>(x, out, cnt, ntiles);
    }
    int start = nblk16 * 16;
    if (start < N) {
        int total = (N - start) * FEAT;
        finalize_rem_kernel<<<(total + 255) / 256, 256, 0, stream>>>(x, out, cnt, start, N);
    }
}